// Pool_20564303414152
// MI455X (gfx1250) — compile-verified
//
#include <hip/hip_runtime.h>

// Problem constants (match the JAX reference)
#define BB   16
#define NN   2048
#define DD   256
#define DOWNK 1024

typedef __attribute__((ext_vector_type(2))) float v2f;
typedef __attribute__((ext_vector_type(8))) float v8f;

// ---------------------------------------------------------------------------
// Kernel 1: scores[b,n] = sigmoid(dot(hidden[b,n,:], W) + bias)
// Uses V_WMMA_F32_16X16X4_F32: each wave computes 16 rows, accumulating 64
// chunks of K=4. B operand is broadcast (every column = W chunk) so every
// column of C holds the scores; lanes 0 and 16 read rows 0-7 / 8-15 per the
// documented 16x16 f32 C layout.
// A 16x4 f32 layout: VGPR0 = K0 (lanes0-15) / K2 (lanes16-31), VGPR1 = K1/K3.
// ---------------------------------------------------------------------------
__global__ __launch_bounds__(256) void scores_wmma_kernel(
    const float* __restrict__ hidden, const float* __restrict__ W,
    const float* __restrict__ bias, float* __restrict__ scores) {
  const int block = blockIdx.x;          // BB * (NN/128) blocks
  const int b     = block / (NN / 128);
  const int rtile = block % (NN / 128);
  const int tid   = threadIdx.x;
  const int wave  = tid >> 5;            // 8 waves * 16 rows = 128 rows/block
  const int lane  = tid & 31;
  const int half  = lane >> 4;           // 0: K={0,1}, 1: K={2,3}
  const int lr    = lane & 15;           // row within the 16-row tile
  const int base  = rtile * 128 + wave * 16;

  const float* hrow = hidden + ((size_t)b * NN + (size_t)(base + lr)) * DD;

  v8f c = {};
  for (int k = 0; k < DD; k += 4) {
    const int kk = k + 2 * half;
    v2f a; a.x = hrow[kk];  a.y = hrow[kk + 1];
    v2f w; w.x = W[kk];     w.y = W[kk + 1];     // broadcast across all 16 cols
    c = __builtin_amdgcn_wmma_f32_16x16x4_f32(
        /*neg_a=*/false, a, /*neg_b=*/false, w,
        /*c_mod=*/(short)0, c, /*reuse_a=*/false, /*reuse_b=*/false);
  }

  // C layout: VGPR j, lanes0-15 -> M=j (N=lane); lanes16-31 -> M=8+j.
  if (lr == 0) {
    const float bv = bias[0];
    float* sout = scores + b * NN + base + half * 8;
#pragma unroll
    for (int j = 0; j < 8; ++j) {
      const float x = c[j] + bv;
      sout[j] = 1.0f / (1.0f + __expf(-x));
    }
  }
}

// ---------------------------------------------------------------------------
// Kernel 2: exact per-batch bitonic sort of 2048 (score desc, index asc ties),
// emit first DOWNK (indices + values). One 1024-thread block per batch,
// 16 KB LDS (well under 320 KB/WGP).
// ---------------------------------------------------------------------------
__global__ __launch_bounds__(1024) void topk_sort_kernel(
    const float* __restrict__ scores, int* __restrict__ sidx,
    float* __restrict__ svals) {
  __shared__ float s[NN];
  __shared__ int   id[NN];
  const int b = blockIdx.x;
  const int t = threadIdx.x;

  s[t]        = scores[b * NN + t];        id[t]        = t;
  s[t + 1024] = scores[b * NN + t + 1024]; id[t + 1024] = t + 1024;
  __syncthreads();

  for (int k = 2; k <= NN; k <<= 1) {
    for (int j = k >> 1; j > 0; j >>= 1) {
#pragma unroll
      for (int which = 0; which < 2; ++which) {
        const int i = t + which * 1024;
        const int l = i ^ j;
        if (l > i) {
          const bool up = ((i & k) == 0);  // region ascending in final order
          const float si = s[i], sl = s[l];
          const int   ii = id[i], il = id[l];
          // "a before b" in final order: higher score, or equal & lower index
          const bool l_first = (sl > si) || (sl == si && il < ii);
          const bool i_first = (si > sl) || (si == sl && ii < il);
          if (up ? l_first : i_first) {
            s[i] = sl; s[l] = si; id[i] = il; id[l] = ii;
          }
        }
        __syncthreads();
      }
    }
  }
  if (t < DOWNK) {
    sidx [b * DOWNK + t] = id[t];
    svals[b * DOWNK + t] = s[t];
  }
}

// ---------------------------------------------------------------------------
// Kernel 3: h[b,i,d] = hidden[b, idx[b,i], d] * vals[b,i]   (coalesced)
// ---------------------------------------------------------------------------
__global__ __launch_bounds__(256) void h_gather_kernel(
    const float* __restrict__ hidden, const int* __restrict__ sidx,
    const float* __restrict__ svals, float* __restrict__ hout) {
  const size_t e = (size_t)blockIdx.x * 256 + threadIdx.x;
  const int d = (int)(e & (DD - 1));
  const int i = (int)((e >> 8) & (DOWNK - 1));
  const int b = (int)(e >> 18);
  const int r = sidx[b * DOWNK + i];
  const float v = svals[b * DOWNK + i];
  hout[e] = hidden[((size_t)b * NN + (size_t)r) * DD + d] * v;
}

// ---------------------------------------------------------------------------
// Kernel 4: a[b,i,j] = adj[b, idx[b,j], idx[b,i]] / deg[b,i]
// One block per output row (b,i): gathered reads hit L2 (the 1024 touched adj
// rows per batch are only 8 MB, trivially L2-resident at 192 MB), stores are
// fully coalesced, and deg + normalize are fused so adj is gathered once.
// ---------------------------------------------------------------------------
__global__ __launch_bounds__(256) void a_norm_kernel(
    const float* __restrict__ adj, const int* __restrict__ sidx,
    float* __restrict__ aout) {
  const int b = blockIdx.x >> 10;
  const int i = blockIdx.x & (DOWNK - 1);
  const int t = threadIdx.x;

  __shared__ float red[256];
  __shared__ float rdeg_sh;

  const int col = sidx[b * DOWNK + i];
  const float* adjb = adj + (size_t)b * NN * NN;
  const int* rows = sidx + b * DOWNK;

  float v[4];
  float acc = 0.0f;
#pragma unroll
  for (int q = 0; q < 4; ++q) {
    const int j = t + q * 256;
    const int r = rows[j];
    v[q] = adjb[(size_t)r * NN + col];
    acc += v[q];
  }

  red[t] = acc;
  __syncthreads();
  for (int s2 = 128; s2 > 0; s2 >>= 1) {
    if (t < s2) red[t] += red[t + s2];
    __syncthreads();
  }
  if (t == 0) rdeg_sh = 1.0f / red[0];
  __syncthreads();
  const float rdeg = rdeg_sh;

  float* out = aout + ((size_t)b * DOWNK + i) * DOWNK;
#pragma unroll
  for (int q = 0; q < 4; ++q) out[t + q * 256] = v[q] * rdeg;
}

// ---------------------------------------------------------------------------
extern "C" void kernel_launch(void* const* d_in, const int* in_sizes, int n_in,
                              void* d_out, int out_size, void* d_ws,
                              size_t ws_size, hipStream_t stream) {
  const float* adj    = (const float*)d_in[0];  // [16,2048,2048]
  const float* hidden = (const float*)d_in[1];  // [16,2048,256]
  const float* W      = (const float*)d_in[2];  // [256,1]
  const float* bias   = (const float*)d_in[3];  // [1]

  float* out_a = (float*)d_out;                             // [16,1024,1024]
  float* out_h = out_a + (size_t)BB * DOWNK * DOWNK;        // [16,1024,256]

  // Workspace layout (256 KB total, tiny vs any plausible ws_size)
  char* ws = (char*)d_ws;
  float* scores = (float*)ws;                                        // 16*2048 f32
  int*   sidx   = (int*)(ws + (size_t)BB * NN * sizeof(float));      // 16*1024 i32
  float* svals  = (float*)(ws + (size_t)BB * NN * sizeof(float)
                              + (size_t)BB * DOWNK * sizeof(int));   // 16*1024 f32

  scores_wmma_kernel<<<BB * (NN / 128), 256, 0, stream>>>(hidden, W, bias, scores);
  topk_sort_kernel<<<BB, 1024, 0, stream>>>(scores, sidx, svals);
  h_gather_kernel<<<(BB * DOWNK * DD) / 256, 256, 0, stream>>>(hidden, sidx, svals, out_h);
  a_norm_kernel<<<BB * DOWNK, 256, 0, stream>>>(adj, sidx, out_a);
}